// SemanticAnchorLoss_57458072486493
// MI455X (gfx1250) — compile-verified
//
#include <hip/hip_runtime.h>

typedef __attribute__((ext_vector_type(2))) float v2f;
typedef __attribute__((ext_vector_type(8))) float v8f;

#define ANCHOR_EPS    1e-8f
#define ANCHOR_WEIGHT 0.01f
#define ANCHOR_D      1024   // embedding dim (reference: 1024)

// Select element `s` (0..7) of an 8-wide accumulator fragment without scratch.
__device__ __forceinline__ float pick8(v8f v, int s) {
    float r = v[0];
#pragma unroll
    for (int i = 1; i < 8; ++i) r = (s == i) ? v[i] : r;
    return r;
}

// One wave32 handles 16 anchor pairs. For each K-chunk of 4 dims we do three
// V_WMMA_F32_16X16X4_F32 accumulations whose diagonals give dot(a,m), ||a||^2,
// ||m||^2 for all 16 pairs simultaneously. A- and B-fragment per-lane layouts
// coincide for f32 16x16x4, so one 8-byte load per lane feeds both operands.
__global__ __launch_bounds__(128) void semantic_anchor_wmma_kernel(
    const float* __restrict__ W,      // [vocab, D] f32
    const int*   __restrict__ idxA,   // [nPairs] i32
    const int*   __restrict__ idxM,   // [nPairs] i32
    float*       __restrict__ terms,  // [nPairs] f32 out: (1 - cos_sim)
    int nPairs)
{
    const int lane        = threadIdx.x & 31;
    const int waveInBlock = threadIdx.x >> 5;
    const int waveId      = blockIdx.x * (blockDim.x >> 5) + waveInBlock;
    const int pairBase    = waveId * 16;
    if (pairBase >= nPairs) return;   // wave-uniform: EXEC stays all-ones

    const int sub   = lane & 15;      // pair (M/N index) this lane loads for
    const int khalf = lane >> 4;      // 0 -> K=0,1 ; 1 -> K=2,3 within chunk

    int p = pairBase + sub;
    if (p >= nPairs) p = nPairs - 1;  // tail clamp (unused for nPairs%16==0)

    const long long ra = (long long)idxA[p];
    const long long rm = (long long)idxM[p];
    const float* aRow = W + ra * (long long)ANCHOR_D + 2 * khalf;
    const float* mRow = W + rm * (long long)ANCHOR_D + 2 * khalf;

    v8f dotAcc = {};  // D[i][j] = sum_k a_i[k] * m_j[k]
    v8f aaAcc  = {};  // D[i][j] = sum_k a_i[k] * a_j[k]
    v8f mmAcc  = {};  // D[i][j] = sum_k m_i[k] * m_j[k]

#pragma unroll 4
    for (int c = 0; c < ANCHOR_D / 4; ++c) {
        v2f av = *(const v2f*)(aRow + 4 * c);
        v2f mv = *(const v2f*)(mRow + 4 * c);
        dotAcc = __builtin_amdgcn_wmma_f32_16x16x4_f32(
            false, av, false, mv, (short)0, dotAcc, false, false);
        aaAcc  = __builtin_amdgcn_wmma_f32_16x16x4_f32(
            false, av, false, av, (short)0, aaAcc, false, false);
        mmAcc  = __builtin_amdgcn_wmma_f32_16x16x4_f32(
            false, mv, false, mv, (short)0, mmAcc, false, false);
    }

    // Diagonal (i,i): i<8 -> lane i, vgpr i ; i>=8 -> lane i+16, vgpr i-8.
    int sel = -1, dpair = -1;
    if (lane < 8)        { sel = lane;      dpair = lane;      }
    else if (lane >= 24) { sel = lane - 24; dpair = lane - 16; }

    if (sel >= 0 && (pairBase + dpair) < nPairs) {
        float dot = pick8(dotAcc, sel);
        float aa  = pick8(aaAcc,  sel);
        float mm  = pick8(mmAcc,  sel);
        float na  = fmaxf(sqrtf(aa), ANCHOR_EPS);
        float nm  = fmaxf(sqrtf(mm), ANCHOR_EPS);
        terms[pairBase + dpair] = 1.0f - dot / (na * nm);
    }
}

// Deterministic single-block final reduction: out = WEIGHT * mean(terms).
__global__ __launch_bounds__(256) void semantic_anchor_reduce_kernel(
    const float* __restrict__ terms, float* __restrict__ out, int n)
{
    __shared__ float sdata[256];
    float s = 0.0f;
    for (int i = threadIdx.x; i < n; i += 256) s += terms[i];
    sdata[threadIdx.x] = s;
    __syncthreads();
#pragma unroll
    for (int off = 128; off > 0; off >>= 1) {
        if ((int)threadIdx.x < off) sdata[threadIdx.x] += sdata[threadIdx.x + off];
        __syncthreads();
    }
    if (threadIdx.x == 0) out[0] = ANCHOR_WEIGHT * sdata[0] / (float)n;
}

extern "C" void kernel_launch(void* const* d_in, const int* in_sizes, int n_in,
                              void* d_out, int out_size, void* d_ws, size_t ws_size,
                              hipStream_t stream) {
    const float* W    = (const float*)d_in[0];
    const int*   idxA = (const int*)d_in[1];
    const int*   idxM = (const int*)d_in[2];
    float*       out  = (float*)d_out;
    float*       terms = (float*)d_ws;   // nPairs floats of scratch

    const int nPairs = in_sizes[1];
    const int waves  = (nPairs + 15) / 16;
    const int blocks = (waves + 3) / 4;  // 128 threads = 4 wave32 per block

    semantic_anchor_wmma_kernel<<<blocks, 128, 0, stream>>>(W, idxA, idxM, terms, nPairs);
    semantic_anchor_reduce_kernel<<<1, 256, 0, stream>>>(terms, out, nPairs);
}